// SSIMLoss_22256520528019
// MI455X (gfx1250) — compile-verified
//
#include <hip/hip_runtime.h>
#include <stdint.h>

// SSIM (32x1x512x512 fp32, 11x11 Gaussian sigma=1.5) via separable conv on
// V_WMMA_F32_16X16X4_F32, with global->LDS staging through the CDNA5 async
// engine (GLOBAL_LOAD_ASYNC_TO_LDS_B128 + s_wait_asynccnt).
// One wave32 per 16x16 output tile. Reference's ssim_map only consumes
// mu1, mu2 and the covariance -> 3 convolutions.

#define IMG_HW 512
#define WAVES_PER_BLOCK 8
#define NTILES (32 * 32 * 32)             // 32 imgs * (512/16)^2 = 32768
#define REG_ROWS 32
#define REG_COLS 36                       // cols x0-8 .. x0+27 (16B aligned base)
#define REG_FLOATS (REG_ROWS * REG_COLS)  // 1152 floats per image
#define WBUF_FLOATS (2 * REG_FLOATS)      // 2304 floats/wave (H overlay needs 1536)

typedef float v2f __attribute__((ext_vector_type(2)));
typedef float v8f __attribute__((ext_vector_type(8)));

// Normalized 11-tap Gaussian (sigma=1.5), g[t] for t in [0,10]; 0 outside band.
__device__ __forceinline__ float gband(int d) {
    switch (d) {
        case 0: case 10: return 0.0010284f;
        case 1: case 9:  return 0.0075988f;
        case 2: case 8:  return 0.0360008f;
        case 3: case 7:  return 0.1093606f;
        case 4: case 6:  return 0.2130055f;
        case 5:          return 0.2660122f;
        default:         return 0.0f;
    }
}

__global__ __launch_bounds__(256)
void ssim_tile_kernel(const float* __restrict__ gt, const float* __restrict__ pr,
                      float* __restrict__ partial) {
    extern __shared__ float smem[];       // WAVES_PER_BLOCK * WBUF_FLOATS floats

    const int tid  = threadIdx.x;
    const int wv   = tid >> 5;
    const int lane = tid & 31;
    const int half = lane >> 4;           // lane half-group
    const int mc   = lane & 15;           // A-row (M) / B-col (N) for this lane
    float* wbuf = smem + wv * WBUF_FLOATS;

    const int tile = blockIdx.x * WAVES_PER_BLOCK + wv;
    const int n    = tile >> 10;
    const int rem  = tile & 1023;
    const int y0   = (rem >> 5) << 4;
    const int x0   = (rem & 31) << 4;
    const int yr0  = y0 - 5;              // image row of region row 0
    const int xr0  = x0 - 8;              // image col of region col 0

    // -------- stage 32x36 region of both images into LDS --------
    const bool interior = (y0 >= 16) && (y0 <= 480) && (x0 >= 16) && (x0 <= 480);
    if (interior) {
        // Async engine path: 9 x b128 per image, lanes contiguous within rows.
        const uint32_t lbase = (uint32_t)(uintptr_t)wbuf;
#pragma unroll
        for (int img = 0; img < 2; ++img) {
            const float* __restrict__ src = img ? pr : gt;
#pragma unroll
            for (int i = 0; i < 9; ++i) {
                const unsigned f   = (unsigned)(i * 32 + lane);   // 0..287 float4s
                const unsigned row = f / 9u;
                const unsigned c4  = f - row * 9u;
                const float* p = src + (((size_t)n << 18) +
                                        ((size_t)(yr0 + (int)row) << 9) +
                                        (size_t)(xr0 + (int)(c4 << 2)));
                const uint32_t laddr = lbase + (uint32_t)img * (REG_FLOATS * 4u) + f * 16u;
                asm volatile("global_load_async_to_lds_b128 %0, %1, off"
                             :: "v"(laddr), "v"(p) : "memory");
            }
        }
        asm volatile("s_wait_asynccnt 0" ::: "memory");
    } else {
        // Border tiles: branchless clamped loads + select-zero (reference zero pad).
#pragma unroll 4
        for (int i = 0; i < 36; ++i) {
            const int f   = i * 32 + lane;        // 0..1151
            const int row = f / 36;
            const int col = f - row * 36;
            const int y = yr0 + row, x = xr0 + col;
            const bool inb = ((unsigned)y < (unsigned)IMG_HW) &&
                             ((unsigned)x < (unsigned)IMG_HW);
            const int yc = inb ? y : 0;
            const int xc = inb ? x : 0;
            const size_t ofs = ((size_t)n << 18) + ((size_t)yc << 9) + (size_t)xc;
            const float va = gt[ofs];
            const float vb = pr[ofs];
            wbuf[f]              = inb ? va : 0.0f;
            wbuf[REG_FLOATS + f] = inb ? vb : 0.0f;
        }
    }

    // Banded Gaussian Toeplitz fragment; serves as horizontal-B and vertical-A.
    v2f wband[7];
#pragma unroll
    for (int k = 0; k < 7; ++k) {
        const int j = 4 * k + 2 * half;
        wband[k].x = gband(j - mc);
        wband[k].y = gband(j + 1 - mc);
    }

    // -------- pull all A-fragments into registers (stride-36 rows: bank-conflict free)
    float a[2][14], b[2][14];
#pragma unroll
    for (int rb = 0; rb < 2; ++rb) {
        const int rbase = (rb * 16 + mc) * REG_COLS;
#pragma unroll
        for (int k = 0; k < 7; ++k) {
            const int col = 3 + 4 * k + 2 * half;   // region col of A-index j
            a[rb][2 * k]     = wbuf[rbase + col];
            a[rb][2 * k + 1] = wbuf[rbase + col + 1];
            b[rb][2 * k]     = wbuf[REG_FLOATS + rbase + col];
            b[rb][2 * k + 1] = wbuf[REG_FLOATS + rbase + col + 1];
        }
    }

    // -------- horizontal pass: H overlays the now-dead input region in LDS --------
    // H layout: wbuf[s*512 + row*16 + col], rows 0..31 (rows >=26 have zero weight).
#pragma unroll
    for (int rb = 0; rb < 2; ++rb) {
#pragma unroll
        for (int s = 0; s < 3; ++s) {
            v8f acc = {};
#pragma unroll
            for (int k = 0; k < 7; ++k) {
                float v0 = a[rb][2 * k], v1 = a[rb][2 * k + 1];
                if (s == 1) { v0 = b[rb][2 * k];                 v1 = b[rb][2 * k + 1]; }
                if (s == 2) { v0 = a[rb][2 * k] * b[rb][2 * k];  v1 = a[rb][2 * k + 1] * b[rb][2 * k + 1]; }
                v2f A; A.x = v0; A.y = v1;
                acc = __builtin_amdgcn_wmma_f32_16x16x4_f32(
                        false, A, false, wband[k], (short)0, acc, false, false);
            }
#pragma unroll
            for (int v = 0; v < 8; ++v)
                wbuf[s * 512 + (rb * 16 + half * 8 + v) * 16 + mc] = acc[v];
        }
    }

    // -------- vertical pass --------
    v8f o[3];
#pragma unroll
    for (int s = 0; s < 3; ++s) {
        v8f acc = {};
#pragma unroll
        for (int k = 0; k < 7; ++k) {
            const int j = 4 * k + 2 * half;
            v2f B;
            B.x = wbuf[s * 512 + j * 16 + mc];
            B.y = wbuf[s * 512 + (j + 1) * 16 + mc];
            acc = __builtin_amdgcn_wmma_f32_16x16x4_f32(
                    false, wband[k], false, B, (short)0, acc, false, false);
        }
        o[s] = acc;
    }

    // -------- elementwise SSIM + wave reduction --------
    const float C1 = 1e-4f;   // 0.01^2
    const float C2 = 9e-4f;   // 0.03^2
    float sum = 0.0f;
#pragma unroll
    for (int v = 0; v < 8; ++v) {
        const float mu1 = o[0][v], mu2 = o[1][v], eab = o[2][v];
        const float m12 = mu1 * mu2;
        const float cov = eab - m12;
        const float msq = mu1 * mu1 + mu2 * mu2;   // reference uses msq in BOTH factors
        const float num = (2.0f * m12 + C1) * (2.0f * cov + C2);
        const float den = (msq + C1) * (msq + C2);
        sum += num / den;
    }
#pragma unroll
    for (int off = 16; off >= 1; off >>= 1)
        sum += __shfl_xor(sum, off, 32);
    if (lane == 0) partial[tile] = sum;
}

// Deterministic fixed-order final reduction (no float atomics).
__global__ __launch_bounds__(256)
void ssim_reduce_kernel(const float* __restrict__ ws, float* __restrict__ out,
                        int n, float scale) {
    __shared__ float sm[256];
    float s = 0.0f;
    for (int i = threadIdx.x; i < n; i += 256) s += ws[i];
    sm[threadIdx.x] = s;
    __syncthreads();
    for (int off = 128; off > 0; off >>= 1) {
        if ((int)threadIdx.x < off) sm[threadIdx.x] += sm[threadIdx.x + off];
        __syncthreads();
    }
    if (threadIdx.x == 0) out[0] = sm[0] * scale;
}

extern "C" void kernel_launch(void* const* d_in, const int* in_sizes, int n_in,
                              void* d_out, int out_size, void* d_ws, size_t ws_size,
                              hipStream_t stream) {
    (void)in_sizes; (void)n_in; (void)out_size; (void)ws_size;
    const float* gt = (const float*)d_in[0];
    const float* pr = (const float*)d_in[1];
    float* out = (float*)d_out;
    float* ws  = (float*)d_ws;   // NTILES floats = 128 KB scratch

    const size_t lds_bytes = (size_t)WAVES_PER_BLOCK * WBUF_FLOATS * sizeof(float); // 72 KB
    ssim_tile_kernel<<<NTILES / WAVES_PER_BLOCK, 256, lds_bytes, stream>>>(gt, pr, ws);
    ssim_reduce_kernel<<<1, 256, 0, stream>>>(ws, out, NTILES,
                                              1.0f / (32.0f * 512.0f * 512.0f));
}